// LightGCNLayer_3367254360245
// MI455X (gfx1250) — compile-verified
//
#include <hip/hip_runtime.h>

// ---------------------------------------------------------------------------
// LightGCN on MI455X (gfx1250, wave32).
//   - SpMM: memory/atomic bound; embeddings (76.8MB each) fit in 192MB L2, so
//     gathers + f32 scatter-atomics are L2-resident. One edge per wave,
//     2 f32 per lane, scalarized edge reads, prefetch of next source row.
//   - Layer accumulate (acc += h): done with chained V_WMMA_F32_16X16X4_F32
//     using one-hot selector A matrices -> bit-exact f32 add on the matrix
//     pipe (free compute on a memory-bound pass).
// ---------------------------------------------------------------------------

typedef float v2f __attribute__((ext_vector_type(2)));
typedef float v8f __attribute__((ext_vector_type(8)));

#define EMBED 64

// ---- init: acc = h = concat(user_emb, item_emb), vectorized float4 --------
__global__ __launch_bounds__(256) void lgcn_init(const float4* __restrict__ user,
                                                 const float4* __restrict__ item,
                                                 float4* __restrict__ acc,
                                                 float4* __restrict__ h,
                                                 long long nUser4, long long nTot4) {
    long long i = (long long)blockIdx.x * blockDim.x + threadIdx.x;
    if (i >= nTot4) return;
    float4 v = (i < nUser4) ? user[i] : item[i - nUser4];
    acc[i] = v;
    h[i]   = v;
}

// ---- zero fill ------------------------------------------------------------
__global__ __launch_bounds__(256) void lgcn_fill0(float4* __restrict__ p, long long n4) {
    long long i = (long long)blockIdx.x * blockDim.x + threadIdx.x;
    if (i >= n4) return;
    p[i] = make_float4(0.f, 0.f, 0.f, 0.f);
}

// ---- COO SpMM with f32 scatter atomics: y[r] += v * x[c] ------------------
// One edge per wave; lane covers 2 consecutive features (global_load_b64 +
// two non-returning global_atomic_add_f32). Edge index is wave-uniform and
// forced into SGPRs so rows/cols/vals come through the scalar path.
__global__ __launch_bounds__(256) void lgcn_spmm(const int*   __restrict__ rows,
                                                 const int*   __restrict__ cols,
                                                 const float* __restrict__ vals,
                                                 const float* __restrict__ x,
                                                 float*       __restrict__ y,
                                                 int nEdges, int nWaves) {
    const int lane = threadIdx.x & 31;
    const int wave = (int)((blockIdx.x * blockDim.x + threadIdx.x) >> 5);
    const int fo   = lane * 2;

    for (int e = wave; e < nEdges; e += nWaves) {
        const int es = __builtin_amdgcn_readfirstlane(e);
        const int r  = rows[es];
        const int c  = cols[es];
        const float v = vals[es];

        // Prefetch next iteration's source row (global_prefetch on gfx1250).
        const int en = es + nWaves;
        if (en < nEdges) {
            const int cn = __builtin_amdgcn_readfirstlane(cols[en]);
            __builtin_prefetch(x + (size_t)cn * EMBED + fo, 0, 3);
        }

        const float2 xv = *reinterpret_cast<const float2*>(x + (size_t)c * EMBED + fo);
        float* yp = y + (size_t)r * EMBED + fo;
        atomicAdd(yp,     v * xv.x);   // global_atomic_add_f32 (no return)
        atomicAdd(yp + 1, v * xv.y);
    }
}

// ---- acc += h via chained V_WMMA_F32_16X16X4_F32 --------------------------
// Tile = 16 nodes x 16 features, one tile per wave.
//   D = sum_{j=0..3} A_j x B_j + C,  A_j[m,k] = delta(m, 4j+k)  (one-hot)
//   B_j[k,n] = h[nb+4j+k, fb+n],     C = acc tile  ->  D = acc + h  (exact)
// Layouts per ISA 7.12.2:
//   A 16x4 : vgpr p, lane l -> (m = l%16, k = p + 2*(l>=16))
//   B 4x16 : vgpr p, lane l -> (k = p + 2*(l>=16), n = l%16)
//   C/D    : vgpr v, lane l -> (m = v + 8*(l>=16), n = l%16)
__global__ __launch_bounds__(256) void lgcn_wmma_add(float* __restrict__ acc,
                                                     const float* __restrict__ h,
                                                     int nTiles, int tilesPerRow) {
    const int lane = threadIdx.x & 31;
    const int tile = (int)(blockIdx.x * 8 + (threadIdx.x >> 5));
    if (tile >= nTiles) return;               // wave-uniform exit; EXEC stays all-1s

    const int nb = (tile / tilesPerRow) * 16; // node base
    const int fb = (tile % tilesPerRow) * 16; // feature base
    const int n  = lane & 15;
    const int hi = lane >> 4;                 // 0: lanes 0-15, 1: lanes 16-31

    // Load C (acc tile): 8 strided f32 per lane.
    v8f c;
    const float* accBase = acc + (size_t)(nb + 8 * hi) * EMBED + fb + n;
#pragma unroll
    for (int vv = 0; vv < 8; ++vv) c[vv] = accBase[(size_t)vv * EMBED];

#pragma unroll
    for (int j = 0; j < 4; ++j) {
        const int m  = lane & 15;
        const int k0 = 2 * hi;
        const int k1 = 2 * hi + 1;
        v2f a, b;
        a.x = (m == 4 * j + k0) ? 1.0f : 0.0f;
        a.y = (m == 4 * j + k1) ? 1.0f : 0.0f;
        const float* hBase = h + (size_t)(nb + 4 * j) * EMBED + fb + n;
        b.x = hBase[(size_t)k0 * EMBED];
        b.y = hBase[(size_t)k1 * EMBED];
        // 8 args: (neg_a, A, neg_b, B, c_mod, C, reuse_a, reuse_b)
        c = __builtin_amdgcn_wmma_f32_16x16x4_f32(false, a, false, b,
                                                  (short)0, c, false, false);
    }

    // Store D back.
    float* outBase = acc + (size_t)(nb + 8 * hi) * EMBED + fb + n;
#pragma unroll
    for (int vv = 0; vv < 8; ++vv) outBase[(size_t)vv * EMBED] = c[vv];
}

// ---- remainder rows (nNodes % 16 != 0; unused for the reference sizes) ----
__global__ __launch_bounds__(256) void lgcn_add_tail(float* __restrict__ acc,
                                                     const float* __restrict__ h,
                                                     long long base, long long count) {
    long long i = (long long)blockIdx.x * blockDim.x + threadIdx.x;
    if (i >= count) return;
    acc[base + i] += h[base + i];
}

// ---- final: acc *= 1/(L+1) -----------------------------------------------
__global__ __launch_bounds__(256) void lgcn_scale(float4* __restrict__ p, long long n4, float s) {
    long long i = (long long)blockIdx.x * blockDim.x + threadIdx.x;
    if (i >= n4) return;
    float4 v = p[i];
    v.x *= s; v.y *= s; v.z *= s; v.w *= s;
    p[i] = v;
}

extern "C" void kernel_launch(void* const* d_in, const int* in_sizes, int n_in,
                              void* d_out, int out_size, void* d_ws, size_t ws_size,
                              hipStream_t stream) {
    const float4* user = (const float4*)d_in[0];
    const float4* item = (const float4*)d_in[1];
    const int*    rows = (const int*)d_in[2];
    const int*    cols = (const int*)d_in[3];
    const float*  vals = (const float*)d_in[4];

    const long long nUsers = in_sizes[0] / EMBED;
    const long long nItems = in_sizes[1] / EMBED;
    const long long nNodes = nUsers + nItems;
    const int       nEdges = in_sizes[2];
    const int       nLayers = 3;   // reference N_LAYERS

    const long long nElem  = nNodes * EMBED;
    const long long nElem4 = nElem / 4;           // EMBED=64 -> divisible by 4
    const long long nUser4 = nUsers * EMBED / 4;

    float* acc = (float*)d_out;                   // running layer sum
    float* hA  = (float*)d_ws;                    // current layer emb
    float* hB  = hA + nElem;                      // next layer emb (ping-pong)

    const int TB = 256;
    const unsigned gElem4 = (unsigned)((nElem4 + TB - 1) / TB);

    // acc = hA = concat(user, item)
    lgcn_init<<<gElem4, TB, 0, stream>>>(user, item, (float4*)acc, (float4*)hA,
                                         nUser4, nElem4);

    // SpMM sizing: 8192 blocks * 8 waves = 65536 waves (~147 edges/wave)
    const int spmmBlocks = 8192;
    const int spmmWaves  = spmmBlocks * (TB / 32);

    // WMMA tile-add sizing
    const long long nodes16     = nNodes & ~15LL;
    const int       tilesPerRow = EMBED / 16;                      // 4
    const int       nTiles      = (int)((nodes16 / 16) * tilesPerRow);
    const unsigned  addBlocks   = (unsigned)((nTiles + 7) / 8);    // 8 waves/block
    const long long tailBase    = nodes16 * EMBED;
    const long long tailCount   = (nNodes - nodes16) * EMBED;

    for (int layer = 0; layer < nLayers; ++layer) {
        lgcn_fill0<<<gElem4, TB, 0, stream>>>((float4*)hB, nElem4);
        lgcn_spmm<<<spmmBlocks, TB, 0, stream>>>(rows, cols, vals, hA, hB,
                                                 nEdges, spmmWaves);
        lgcn_wmma_add<<<addBlocks, TB, 0, stream>>>(acc, hB, nTiles, tilesPerRow);
        if (tailCount > 0) {
            lgcn_add_tail<<<(unsigned)((tailCount + TB - 1) / TB), TB, 0, stream>>>(
                acc, hB, tailBase, tailCount);
        }
        float* t = hA; hA = hB; hB = t;           // ping-pong (host-side, pure)
    }

    lgcn_scale<<<gElem4, TB, 0, stream>>>((float4*)acc, nElem4,
                                          1.0f / (float)(nLayers + 1));
}